// SynthMorphLoss_15487652069654
// MI455X (gfx1250) — compile-verified
//
#include <hip/hip_runtime.h>
#include <hip/hip_bf16.h>

typedef __attribute__((ext_vector_type(2))) float v2f;
typedef __attribute__((ext_vector_type(8))) float v8f;

#define NUMC   26
#define D_     160
#define H_     192
#define W_     224
#define PLANE  (H_ * W_)        // 43008
#define VOL    (D_ * PLANE)     // 6881280
#define NCH    3
#define NVF    (NCH * VOL)      // 20643840

// Workspace layout in 32-bit units:
//  [0..25]  fixed-label counts
//  [26..51] moving-label counts
//  [52..77] intersection counts
//  [78..80] float sums: sum(dz^2), sum(dy^2), sum(dx^2)
#define WS_FIX 0
#define WS_MOV 26
#define WS_INT 52
#define WS_SUM 78

__global__ void init_ws_kernel(unsigned int* __restrict__ ws) {
  int i = threadIdx.x;
  if (i < 81) ws[i] = 0u;
}

// Per-wave replicated LDS histograms; fixed-count and intersection-count packed
// into one u64 (low 32 = fixed, high 32 = intersection) -> ds_add_u64.
__global__ void dice_hist_kernel(const int* __restrict__ f,
                                 const int* __restrict__ m,
                                 unsigned int* __restrict__ ws) {
  __shared__ unsigned long long fm[8][NUMC];
  __shared__ unsigned int       mv[8][NUMC];
  const int tid  = threadIdx.x;        // blockDim.x == 256 -> 8 wave32s
  const int wave = tid >> 5;

  for (int i = tid; i < 8 * NUMC; i += 256) {
    (&fm[0][0])[i] = 0ull;
    (&mv[0][0])[i] = 0u;
  }
  __syncthreads();

  const int4* f4 = (const int4*)f;
  const int4* m4 = (const int4*)m;
  const int n4 = VOL / 4;
  const int stride = gridDim.x * blockDim.x;
  for (int i = blockIdx.x * blockDim.x + tid; i < n4; i += stride) {
    int4 a = f4[i];
    int4 b = m4[i];
    atomicAdd(&fm[wave][a.x], 1ull | ((unsigned long long)(a.x == b.x) << 32));
    atomicAdd(&mv[wave][b.x], 1u);
    atomicAdd(&fm[wave][a.y], 1ull | ((unsigned long long)(a.y == b.y) << 32));
    atomicAdd(&mv[wave][b.y], 1u);
    atomicAdd(&fm[wave][a.z], 1ull | ((unsigned long long)(a.z == b.z) << 32));
    atomicAdd(&mv[wave][b.z], 1u);
    atomicAdd(&fm[wave][a.w], 1ull | ((unsigned long long)(a.w == b.w) << 32));
    atomicAdd(&mv[wave][b.w], 1u);
  }
  __syncthreads();

  if (tid < NUMC) {
    unsigned long long accFM = 0ull;
    unsigned int accM = 0u;
    for (int w = 0; w < 8; ++w) { accFM += fm[w][tid]; accM += mv[w][tid]; }
    atomicAdd(&ws[WS_FIX + tid], (unsigned int)(accFM & 0xffffffffull));
    atomicAdd(&ws[WS_INT + tid], (unsigned int)(accFM >> 32));
    atomicAdd(&ws[WS_MOV + tid], accM);
  }
}

// Streaming float4 stencil; neighbor re-loads hit the 192MB L2 (whole field fits),
// so HBM traffic stays ~1 pass.
__global__ void diffusion_kernel(const float* __restrict__ vf,
                                 float* __restrict__ sums) {
  const int tid = threadIdx.x;
  const float4* p = (const float4*)vf;
  const int n4 = NVF / 4;
  const int stride = gridDim.x * blockDim.x;
  float sz = 0.f, sy = 0.f, sx = 0.f;

  for (int i = blockIdx.x * blockDim.x + tid; i < n4; i += stride) {
    const int base = i * 4;
    const int r  = base % VOL;
    const int z  = r / PLANE;
    const int r2 = r % PLANE;
    const int y  = r2 / W_;
    const int x  = r2 % W_;

    float4 c = p[i];

    // dx: 3 in-vector terms, plus boundary term needing one scalar neighbor
    float d0 = c.y - c.x, d1 = c.z - c.y, d2 = c.w - c.z;
    sx += d0 * d0 + d1 * d1 + d2 * d2;
    if (x < W_ - 4) {
      float nx = vf[base + 4];
      float d3 = nx - c.w;
      sx += d3 * d3;
    }
    // dy
    if (y < H_ - 1) {
      float4 u = p[i + W_ / 4];
      float e0 = u.x - c.x, e1 = u.y - c.y, e2 = u.z - c.z, e3 = u.w - c.w;
      sy += e0 * e0 + e1 * e1 + e2 * e2 + e3 * e3;
    }
    // dz
    if (z < D_ - 1) {
      float4 u = p[i + PLANE / 4];
      float e0 = u.x - c.x, e1 = u.y - c.y, e2 = u.z - c.z, e3 = u.w - c.w;
      sz += e0 * e0 + e1 * e1 + e2 * e2 + e3 * e3;
    }
  }

  // wave32 butterfly reduction
  for (int o = 16; o > 0; o >>= 1) {
    sz += __shfl_down(sz, o);
    sy += __shfl_down(sy, o);
    sx += __shfl_down(sx, o);
  }
  __shared__ float red[8][3];
  const int wave = tid >> 5, lane = tid & 31;
  if (lane == 0) { red[wave][0] = sz; red[wave][1] = sy; red[wave][2] = sx; }
  __syncthreads();
  if (tid == 0) {
    float tz = 0.f, ty = 0.f, tx = 0.f;
    for (int w = 0; w < 8; ++w) { tz += red[w][0]; ty += red[w][1]; tx += red[w][2]; }
    atomicAdd(&sums[0], tz);
    atomicAdd(&sums[1], ty);
    atomicAdd(&sums[2], tx);
  }
}

// One wave. Sum of the 25 per-class dice values via v_wmma_f32_16x16x4_f32:
// A holds the dice values (zeros elsewhere), B = all ones, so D's rows are
// row-sums of A; lanes 0 and 16 hold D[0..7][0] / D[8..15][0] -> total sum.
__global__ void finalize_kernel(const unsigned int* __restrict__ ws,
                                float* __restrict__ out) {
  const int lane = threadIdx.x; // launched with 32 threads, EXEC all ones
  const float* sums = (const float*)(ws + WS_SUM);

  auto dice_j = [&](int j) -> float {
    if (j < 0 || j > 24) return 0.f;
    const int c = j + 1; // valid classes 1..25 (IGNORE_LABEL == 0)
    float fx = (float)ws[WS_FIX + c];
    float mv = (float)ws[WS_MOV + c];
    float in = (float)ws[WS_INT + c];
    return (2.f * in + 1e-5f) / (fx + mv + 1e-5f);
  };

  float a0 = (lane < 16) ? dice_j(lane)      : 0.f;
  float a1 = (lane < 16) ? dice_j(lane + 16) : 0.f;

  v2f A; A[0] = a0;  A[1] = a1;
  v2f B; B[0] = 1.f; B[1] = 1.f;
  v8f C = {};
  C = __builtin_amdgcn_wmma_f32_16x16x4_f32(
      /*neg_a=*/false, A, /*neg_b=*/false, B,
      /*c_mod=*/(short)0, C, /*reuse_a=*/false, /*reuse_b=*/false);

  float rs = C[0] + C[1] + C[2] + C[3] + C[4] + C[5] + C[6] + C[7];
  float diceSum = __shfl(rs, 0) + __shfl(rs, 16);

  float sim = 1.f - diceSum * (1.f / 25.f);

  // element counts: 3*(D-1)*H*W, 3*D*(H-1)*W, 3*D*H*(W-1)
  float sz = sums[0], sy = sums[1], sx = sums[2];
  float smooth = (sz * (1.f / 20514816.f) +
                  sy * (1.f / 20536320.f) +
                  sx * (1.f / 20551680.f)) * (1.f / 3.f);

  if (lane == 0) {
    out[0] = sim + smooth;
    out[1] = sim;
    out[2] = smooth;
  }
}

extern "C" void kernel_launch(void* const* d_in, const int* in_sizes, int n_in,
                              void* d_out, int out_size, void* d_ws, size_t ws_size,
                              hipStream_t stream) {
  const int*   fixedL  = (const int*)d_in[0];
  const int*   movingL = (const int*)d_in[1];
  const float* vf      = (const float*)d_in[2];
  unsigned int* ws = (unsigned int*)d_ws;
  float* out = (float*)d_out;

  init_ws_kernel<<<1, 128, 0, stream>>>(ws);
  dice_hist_kernel<<<2048, 256, 0, stream>>>(fixedL, movingL, ws);
  diffusion_kernel<<<2048, 256, 0, stream>>>(vf, (float*)(ws + WS_SUM));
  finalize_kernel<<<1, 32, 0, stream>>>(ws, out);
}